// KKAN_Convolutional_Network_19481971654959
// MI455X (gfx1250) — compile-verified
//
#include <hip/hip_runtime.h>
#include <hip/hip_bf16.h>

typedef __attribute__((ext_vector_type(2))) float v2f;
typedef __attribute__((ext_vector_type(8))) float v8f;

// ---------------------------------------------------------------------------
// Uniform-grid Cox-de Boor B-spline bases, order 3.
// grid[g] = x0 + g*h, g = 0..NG-1.  Produces NG-4 bases.
// Fully unrolled so all arrays stay in registers.
// ---------------------------------------------------------------------------
template <int NG>
__device__ __forceinline__ void bspl(float x, float x0, float h, float* out /*NG-4*/) {
    float b[NG - 1];
#pragma unroll
    for (int t = 0; t < NG - 1; ++t) {
        float g0 = x0 + t * h;
        float g1 = x0 + (t + 1) * h;
        b[t] = (x >= g0 && x < g1) ? 1.0f : 0.0f;
    }
#pragma unroll
    for (int j = 1; j <= 3; ++j) {
        float inv = 1.0f / (j * h);   // uniform grid: grid[t+j]-grid[t] == j*h
#pragma unroll
        for (int t = 0; t + j < NG - 1; ++t) {
            float gt  = x0 + t * h;
            float gtj = x0 + (t + j + 1) * h;
            b[t] = (x - gt) * inv * b[t] + (gtj - x) * inv * b[t + 1];
        }
    }
#pragma unroll
    for (int t = 0; t < NG - 4; ++t) out[t] = b[t];
}

__device__ __forceinline__ float silu(float p) { return p / (1.0f + expf(-p)); }

// ---------------------------------------------------------------------------
// KAN 3x3 "conv": 5 independent KANLinear(9->1) applied to each 3x3 patch of
// each input channel.  in: [B,C,H,W] -> out: [B,C*5,H-2,W-2]
// ---------------------------------------------------------------------------
template <int S, int NG>
__global__ void kan_conv_kernel(const float* __restrict__ in, float* __restrict__ out,
                                const float* __restrict__ baseW,   // [5,9]
                                const float* __restrict__ splW,    // [5,9,S]
                                const float* __restrict__ sclW,    // [5,9]
                                int total, int C, int H, int W, float x0, float h) {
    __shared__ float s_bw[45];
    __shared__ float s_sw[45 * S];
    for (int t = threadIdx.x; t < 45; t += blockDim.x) s_bw[t] = baseW[t];
    for (int t = threadIdx.x; t < 45 * S; t += blockDim.x) s_sw[t] = splW[t] * sclW[t / S];
    __syncthreads();

    int idx = blockIdx.x * blockDim.x + threadIdx.x;
    if (idx >= total) return;

    const int Ho = H - 2, Wo = W - 2;
    int ox = idx % Wo;
    int t1 = idx / Wo;
    int oy = t1 % Ho;
    int t2 = t1 / Ho;
    int c  = t2 % C;
    int b  = t2 / C;

    const float* src = in + ((size_t)(b * C + c) * H + oy) * W + ox;

    float acc[5] = {0.f, 0.f, 0.f, 0.f, 0.f};
    for (int i = 0; i < 9; ++i) {                     // patch element (row-major 3x3)
        int dy = i / 3, dx = i % 3;
        float p  = src[dy * W + dx];
        float sl = silu(p);
        float bas[NG - 4];
        bspl<NG>(p, x0, h, bas);
#pragma unroll
        for (int k = 0; k < 5; ++k) {
            float a = sl * s_bw[k * 9 + i];
#pragma unroll
            for (int s2 = 0; s2 < S; ++s2) a += bas[s2] * s_sw[(k * 9 + i) * S + s2];
            acc[k] += a;
        }
    }
    const int Co = C * 5;
#pragma unroll
    for (int k = 0; k < 5; ++k)
        out[((size_t)(b * Co + c * 5 + k) * Ho + oy) * Wo + ox] = acc[k];
}

// ---------------------------------------------------------------------------
// 2x2 max-pool (floor semantics: rows/cols beyond H2*2 dropped)
// in: [N,H,W] -> out: [N,H/2,W/2]
// ---------------------------------------------------------------------------
__global__ void maxpool2_kernel(const float* __restrict__ in, float* __restrict__ out,
                                int N, int H, int W) {
    int H2 = H / 2, W2 = W / 2;
    int total = N * H2 * W2;
    int idx = blockIdx.x * blockDim.x + threadIdx.x;
    if (idx >= total) return;
    int ox = idx % W2;
    int oy = (idx / W2) % H2;
    int n  = idx / (W2 * H2);
    const float* p = in + ((size_t)n * H + oy * 2) * W + ox * 2;
    out[idx] = fmaxf(fmaxf(p[0], p[1]), fmaxf(p[W], p[W + 1]));
}

// ---------------------------------------------------------------------------
// Head: KANLinear(625 -> 10) + log_softmax, via V_WMMA_F32_16X16X4_F32.
// One wave (32 lanes) per tile of 16 samples. For each input feature i, a
// 16(K,pad from 14) x 16(M/N) A/B block is staged in LDS and consumed by
// four f32 WMMAs (K = 4 each), accumulating logits in a v8f C fragment.
// f: [B,625]  baseW: [10,625]  splW: [10,625,13]  sclW: [10,625]  out: [B,10]
// ---------------------------------------------------------------------------
__global__ void kan_head_kernel(const float* __restrict__ f,
                                const float* __restrict__ baseW,
                                const float* __restrict__ splW,
                                const float* __restrict__ sclW,
                                float* __restrict__ out, int B) {
    __shared__ float As[16][16];   // [M(sample)][K(feature)]
    __shared__ float Bs[16][16];   // [K(feature)][N(output)]
    __shared__ float Cs[16][16];   // [M][N] logits staging

    const int lane = threadIdx.x;          // 0..31
    const int row  = lane & 15;
    const int hi   = lane >> 4;            // 0: lanes 0-15, 1: lanes 16-31
    const int m0   = blockIdx.x * 16;

    // zero A/B padding (rows/cols 14,15 of A; unused output cols of B)
    for (int t = lane; t < 256; t += 32) {
        (&As[0][0])[t] = 0.0f;
        (&Bs[0][0])[t] = 0.0f;
    }
    __syncthreads();

    v8f acc = {};
    for (int i = 0; i < 625; ++i) {
        if (hi == 0) {
            // lanes 0-15: features for sample m0+row
            int m = m0 + row;
            float xv = (m < B) ? f[(size_t)m * 625 + i] : 0.0f;
            float bas[13];
            bspl<17>(xv, -0.3f, 0.1f, bas);
            As[row][0] = silu(xv);
#pragma unroll
            for (int s2 = 0; s2 < 13; ++s2) As[row][1 + s2] = bas[s2];
        } else {
            // lanes 16-25: weight column for output o = row
            int o = row;
            if (o < 10) {
                float sc = sclW[(size_t)o * 625 + i];
                Bs[0][o] = baseW[(size_t)o * 625 + i];
#pragma unroll
                for (int s2 = 0; s2 < 13; ++s2)
                    Bs[1 + s2][o] = splW[((size_t)o * 625 + i) * 13 + s2] * sc;
            }
        }
        __syncthreads();

        // 4x V_WMMA_F32_16X16X4_F32 over K = 0..15.
        // A 16x4 layout: lanes 0-15 hold K={4t,4t+1}, lanes 16-31 K={4t+2,4t+3}.
        // B 4x16 layout mirrors it (rows striped across lanes within a VGPR).
#pragma unroll
        for (int t = 0; t < 4; ++t) {
            int k0 = 4 * t + 2 * hi;
            v2f a, bfr;
            a[0]   = As[row][k0];
            a[1]   = As[row][k0 + 1];
            bfr[0] = Bs[k0][row];
            bfr[1] = Bs[k0 + 1][row];
            acc = __builtin_amdgcn_wmma_f32_16x16x4_f32(
                false, a, false, bfr, (short)0, acc, false, false);
        }
        __syncthreads();
    }

    // Spill C fragment: VGPR j holds M = hi*8 + j at column N = row.
#pragma unroll
    for (int j = 0; j < 8; ++j) Cs[hi * 8 + j][row] = acc[j];
    __syncthreads();

    if (hi == 0) {
        int m = m0 + row;
        if (m < B) {
            float mx = Cs[row][0];
#pragma unroll
            for (int o = 1; o < 10; ++o) mx = fmaxf(mx, Cs[row][o]);
            float se = 0.0f;
#pragma unroll
            for (int o = 0; o < 10; ++o) se += expf(Cs[row][o] - mx);
            float lse = mx + logf(se);
#pragma unroll
            for (int o = 0; o < 10; ++o) out[(size_t)m * 10 + o] = Cs[row][o] - lse;
        }
    }
}

// ---------------------------------------------------------------------------
// Launcher
// ---------------------------------------------------------------------------
extern "C" void kernel_launch(void* const* d_in, const int* in_sizes, int n_in,
                              void* d_out, int out_size, void* d_ws, size_t ws_size,
                              hipStream_t stream) {
    const float* x    = (const float*)d_in[0];  // [B,1,28,28]
    const float* c1b  = (const float*)d_in[1];  // [5,9]
    const float* c1s  = (const float*)d_in[2];  // [5,9,8]
    const float* c1sc = (const float*)d_in[3];  // [5,9]
    const float* c2b  = (const float*)d_in[4];  // [5,9]
    const float* c2s  = (const float*)d_in[5];  // [5,9,13]
    const float* c2sc = (const float*)d_in[6];  // [5,9]
    const float* k1b  = (const float*)d_in[7];  // [10,625]
    const float* k1s  = (const float*)d_in[8];  // [10,625,13]
    const float* k1sc = (const float*)d_in[9];  // [10,625]
    float* out = (float*)d_out;
    float* ws  = (float*)d_ws;

    const int B = in_sizes[0] / (28 * 28);

    // workspace layout (floats)
    size_t o_c1 = 0;                                    // [B,5,26,26]
    size_t n_c1 = (size_t)B * 5 * 26 * 26;
    size_t o_p1 = o_c1 + n_c1;                          // [B,5,13,13]
    size_t n_p1 = (size_t)B * 5 * 13 * 13;
    size_t o_c2 = o_p1 + n_p1;                          // [B,25,11,11]
    size_t n_c2 = (size_t)B * 25 * 11 * 11;
    size_t o_p2 = o_c2 + n_c2;                          // [B,625]
    (void)ws_size; (void)n_in; (void)out_size;

    const int TB = 256;

    // conv1: grid_size=5, range [-1,1] -> grid[g] = g*0.4 - 2.2, 12 pts, S=8
    {
        int total = B * 1 * 26 * 26;
        kan_conv_kernel<8, 12><<<(total + TB - 1) / TB, TB, 0, stream>>>(
            x, ws + o_c1, c1b, c1s, c1sc, total, 1, 28, 28, -2.2f, 0.4f);
    }
    // pool1: [B,5,26,26] -> [B,5,13,13]
    {
        int total = B * 5 * 13 * 13;
        maxpool2_kernel<<<(total + TB - 1) / TB, TB, 0, stream>>>(
            ws + o_c1, ws + o_p1, B * 5, 26, 26);
    }
    // conv2: grid_size=10, range [-1,1] -> grid[g] = g*0.2 - 1.6, 17 pts, S=13
    {
        int total = B * 5 * 11 * 11;
        kan_conv_kernel<13, 17><<<(total + TB - 1) / TB, TB, 0, stream>>>(
            ws + o_p1, ws + o_c2, c2b, c2s, c2sc, total, 5, 13, 13, -1.6f, 0.2f);
    }
    // pool2: [B,25,11,11] -> [B,25,5,5]
    {
        int total = B * 25 * 5 * 5;
        maxpool2_kernel<<<(total + TB - 1) / TB, TB, 0, stream>>>(
            ws + o_c2, ws + o_p2, B * 25, 11, 11);
    }
    // head: WMMA GEMM + log_softmax
    {
        int tiles = (B + 15) / 16;
        kan_head_kernel<<<tiles, 32, 0, stream>>>(ws + o_p2, k1b, k1s, k1sc, out, B);
    }
}